// GCNModel_42245298323767
// MI455X (gfx1250) — compile-verified
//
#include <hip/hip_runtime.h>

typedef __attribute__((ext_vector_type(2))) float v2f;
typedef __attribute__((ext_vector_type(8))) float v8f;

#define IN_DIM 128
#define H1 32
#define H2 16

__device__ __forceinline__ void atomAddF(float* p, float v) {
  // native global_atomic_add_f32 (no CAS loop); L2 atomics are fine for f32 on CDNA5
  unsafeAtomicAdd(p, v);
}

__global__ void fill_f32(float* __restrict__ p, float val, int n) {
  int i = blockIdx.x * blockDim.x + threadIdx.x;
  if (i < n) p[i] = val;
}

__global__ void deg_accum(const int* __restrict__ dst, const float* __restrict__ w,
                          float* __restrict__ deg, int nE) {
  int e = blockIdx.x * blockDim.x + threadIdx.x;
  if (e < nE) atomAddF(&deg[dst[e]], w[e]);
}

__global__ void deg_to_dinv(float* __restrict__ deg, int n) {
  int i = blockIdx.x * blockDim.x + threadIdx.x;
  if (i < n) {
    float d = deg[i];
    deg[i] = d > 0.f ? rsqrtf(d) : 0.f;
  }
}

// One wave per 16x16 output tile; f32 WMMA, K stepped by 4.
// A 16x4 layout: lanes 0-15 hold K=0,1 ; lanes 16-31 hold K=2,3 (row M = lane&15)
// B 4x16 layout mirrors A (col N = lane&15)
// D 16x16: VGPR v -> row M = v + 8*(lane>>4), col N = lane&15
template <int OUTD>
__global__ void gemm_wmma_f32(const float* __restrict__ X, const float* __restrict__ W,
                              float* __restrict__ Y, int nrows, int kdim) {
  const int lane = threadIdx.x & 31;
  const int wave = (blockIdx.x * blockDim.x + threadIdx.x) >> 5;
  const int ntiles = OUTD >> 4;
  const int mtiles = nrows >> 4;
  if (wave >= mtiles * ntiles) return;  // wave-uniform: EXEC stays all-1s for WMMA
  const int mt = wave / ntiles;
  const int nt = wave % ntiles;
  const int r  = lane & 15;
  const int hi = lane >> 4;  // 0: K 0-1, 1: K 2-3
  const float* __restrict__ xrow = X + (long long)(mt * 16 + r) * kdim;
  const float* __restrict__ wcol = W + nt * 16 + r;
  v8f acc = {};
  for (int k0 = 0; k0 < kdim; k0 += 4) {
    const int kk = k0 + 2 * hi;
    v2f a, b;
    a.x = xrow[kk];
    a.y = xrow[kk + 1];
    b.x = wcol[(long long)kk * OUTD];
    b.y = wcol[(long long)(kk + 1) * OUTD];
    acc = __builtin_amdgcn_wmma_f32_16x16x4_f32(false, a, false, b, (short)0, acc,
                                                false, false);
  }
  float* __restrict__ yrow = Y + (long long)(mt * 16 + hi * 8) * OUTD + nt * 16 + r;
#pragma unroll
  for (int v = 0; v < 8; v++) yrow[(long long)v * OUTD] = acc[v];
}

// Layer-1 aggregation: wave handles 8 edges, lane == feature (32 feats == wave32).
// Gathers xw[src*32..+31] (128B coalesced) and scatters coalesced f32 atomics.
__global__ void agg_edges_32(const int* __restrict__ src, const int* __restrict__ dst,
                             const float* __restrict__ w, const float* __restrict__ dinv,
                             const float* __restrict__ xw, float* __restrict__ agg, int nE) {
  const int lane = threadIdx.x & 31;
  const int wave = (blockIdx.x * blockDim.x + threadIdx.x) >> 5;
  const int e0 = wave * 8;
#pragma unroll
  for (int i = 0; i < 8; i++) {
    int e = e0 + i;
    if (e >= nE) break;
    int s = src[e], d = dst[e];
    float nn = dinv[s] * w[e] * dinv[d];
    atomAddF(&agg[(long long)d * 32 + lane], nn * xw[(long long)s * 32 + lane]);
  }
}

// Layer-2 aggregation: 16 feats -> lanes 0-15 handle edge 2i, lanes 16-31 edge 2i+1.
__global__ void agg_edges_16(const int* __restrict__ src, const int* __restrict__ dst,
                             const float* __restrict__ w, const float* __restrict__ dinv,
                             const float* __restrict__ xw, float* __restrict__ agg, int nE) {
  const int lane = threadIdx.x & 31;
  const int wave = (blockIdx.x * blockDim.x + threadIdx.x) >> 5;
  const int sub = lane >> 4;
  const int f = lane & 15;
  const int e0 = wave * 16;
#pragma unroll
  for (int i = 0; i < 8; i++) {
    int e = e0 + 2 * i + sub;
    if (e < nE) {
      int s = src[e], d = dst[e];
      float nn = dinv[s] * w[e] * dinv[d];
      atomAddF(&agg[(long long)d * 16 + f], nn * xw[(long long)s * 16 + f]);
    }
  }
}

__global__ void finalize_relu_32(float* __restrict__ agg, const float* __restrict__ xw,
                                 const float* __restrict__ dinv,
                                 const float* __restrict__ bias, int n) {
  int idx = blockIdx.x * blockDim.x + threadIdx.x;
  if (idx < n) {
    int i = idx >> 5, f = idx & 31;
    float di = dinv[i];
    float v = agg[idx] + di * di * xw[idx] + bias[f];
    agg[idx] = v > 0.f ? v : 0.f;  // agg becomes h in place
  }
}

__global__ void finalize_16(float* __restrict__ out, const float* __restrict__ xw,
                            const float* __restrict__ dinv,
                            const float* __restrict__ bias, int n) {
  int idx = blockIdx.x * blockDim.x + threadIdx.x;
  if (idx < n) {
    int i = idx >> 4, f = idx & 15;
    float di = dinv[i];
    out[idx] = out[idx] + di * di * xw[idx] + bias[f];
  }
}

extern "C" void kernel_launch(void* const* d_in, const int* in_sizes, int n_in,
                              void* d_out, int out_size, void* d_ws, size_t ws_size,
                              hipStream_t stream) {
  const float* x  = (const float*)d_in[0];
  const int*   ei = (const int*)d_in[1];   // [2, E] flat: src then dst
  const float* w  = (const float*)d_in[2];
  const float* W1 = (const float*)d_in[3];
  const float* b1 = (const float*)d_in[4];
  const float* W2 = (const float*)d_in[5];
  const float* b2 = (const float*)d_in[6];

  const int nN = in_sizes[0] / IN_DIM;  // 100000 (divisible by 16)
  const int nE = in_sizes[2];           // 3200000
  const int* src = ei;
  const int* dst = ei + nE;

  // workspace layout (floats): dinv[N] | xw1[N*32] | agg1/h[N*32] | xw2[N*16]
  float* ws   = (float*)d_ws;
  float* dinv = ws;
  float* xw1  = dinv + nN;
  float* agg1 = xw1 + (long long)nN * H1;
  float* xw2  = agg1 + (long long)nN * H1;
  float* out  = (float*)d_out;

  const int T = 256;
  auto blk = [](long long n, int t) { return (int)((n + t - 1) / t); };

  // degree (self-loop weight 1) -> dinv in place
  fill_f32<<<blk(nN, T), T, 0, stream>>>(dinv, 1.0f, nN);
  deg_accum<<<blk(nE, T), T, 0, stream>>>(dst, w, dinv, nE);
  deg_to_dinv<<<blk(nN, T), T, 0, stream>>>(dinv, nN);

  // layer 1: xw1 = x @ W1 via f32 WMMA (one wave per 16x16 tile)
  {
    long long tiles = (long long)(nN / 16) * (H1 / 16);
    gemm_wmma_f32<H1><<<blk(tiles * 32, 128), 128, 0, stream>>>(x, W1, xw1, nN, IN_DIM);
  }
  fill_f32<<<blk((long long)nN * H1, T), T, 0, stream>>>(agg1, 0.0f, nN * H1);
  {
    long long waves = (nE + 7) / 8;
    agg_edges_32<<<blk(waves * 32, T), T, 0, stream>>>(src, dst, w, dinv, xw1, agg1, nE);
  }
  finalize_relu_32<<<blk((long long)nN * H1, T), T, 0, stream>>>(agg1, xw1, dinv, b1,
                                                                 nN * H1);

  // layer 2: xw2 = h @ W2 via f32 WMMA
  {
    long long tiles = (long long)(nN / 16) * (H2 / 16);
    gemm_wmma_f32<H2><<<blk(tiles * 32, 128), 128, 0, stream>>>(agg1, W2, xw2, nN, H1);
  }
  fill_f32<<<blk((long long)nN * H2, T), T, 0, stream>>>(out, 0.0f, nN * H2);
  {
    long long waves = (nE + 15) / 16;
    agg_edges_16<<<blk(waves * 32, T), T, 0, stream>>>(src, dst, w, dinv, xw2, out, nE);
  }
  finalize_16<<<blk((long long)nN * H2, T), T, 0, stream>>>(out, xw2, dinv, b2, nN * H2);
}